// AdaptedCrossEntropySurvivalLoss_89962384982548
// MI455X (gfx1250) — compile-verified
//
#include <hip/hip_runtime.h>

#define EPSV 1e-7f
#define ONE_M_EPS (1.0f - 1e-7f)

typedef __attribute__((ext_vector_type(2))) float v2f;
typedef __attribute__((ext_vector_type(8))) float v8f;

// Sum 32 per-lane f32 values across the wave using V_WMMA_F32_16X16X4_F32.
// A[m,0]=acc(lane m), A[m,2]=acc(lane m+16), A[m,1]=A[m,3]=0; B = ones(4x16)
// => D[m,n] = acc[m]+acc[m+16]. Sum the 8 D VGPRs per lane, then fold the
// two half-wave partials with one shfl_xor(16). Result valid in all lanes.
__device__ __forceinline__ float wave_sum32_wmma(float acc) {
    v2f a; a[0] = acc;  a[1] = 0.0f;
    v2f b; b[0] = 1.0f; b[1] = 1.0f;
    v8f c = {};
    c = __builtin_amdgcn_wmma_f32_16x16x4_f32(
            /*neg_a=*/false, a, /*neg_b=*/false, b,
            /*c_mod=*/(short)0, c, /*reuse_a=*/false, /*reuse_b=*/false);
    float s = ((c[0] + c[1]) + (c[2] + c[3])) + ((c[4] + c[5]) + (c[6] + c[7]));
    s += __shfl_xor(s, 16, 32);
    return s;
}

// Per-row setup: bitmask of active columns + affine transform selecting h vs 1-h.
//   event==0 (censored): active cols [0, t),   x = 1 - h   -> sgn=-1, off=1
//   event!=0 (event):    active cols [t-1,32), x = h       -> sgn=+1, off=0
__device__ __forceinline__ void row_params(int t_raw, int ev,
                                           unsigned& mask, float& sgn, float& off) {
    int t = min(max(t_raw, 1), 32);
    bool isEvt = (ev != 0);
    mask = isEvt ? (0xFFFFFFFFu << (t - 1)) : (0xFFFFFFFFu >> (32 - t));
    sgn  = isEvt ? 1.0f : -1.0f;
    off  = isEvt ? 0.0f : 1.0f;
}

__global__ __launch_bounds__(256) void survloss_partial_kernel(
    const float* __restrict__ preds, const int* __restrict__ targets,
    float* __restrict__ partial, int n_rows, int n_groups) {
    const int lane      = threadIdx.x & 31;
    const int waveInBlk = threadIdx.x >> 5;
    const int gwave     = (blockIdx.x * blockDim.x + threadIdx.x) >> 5;
    const int nwaves    = (gridDim.x * blockDim.x) >> 5;

    const float4* __restrict__ preds4 = (const float4*)preds;
    const int2*   __restrict__ tgt2   = (const int2*)targets;

    float acc = 0.0f;

    // Main loop: each wave iteration streams 4 rows (512 B) via float4 loads.
    for (int g = gwave; g < n_groups; g += nwaves) {
        // Speculative prefetch of the next chunk (global_prefetch_b8).
        __builtin_prefetch(&preds4[(size_t)(g + nwaves) * 32 + lane], 0, 1);

        float4 h4 = preds4[(size_t)g * 32 + lane];
        int row   = g * 4 + (lane >> 3);
        int2 te   = tgt2[row];

        unsigned mask; float sgn, off;
        row_params(te.x, te.y, mask, sgn, off);

        const int col0 = (lane & 7) * 4;
        float hv[4] = {h4.x, h4.y, h4.z, h4.w};
#pragma unroll
        for (int cc = 0; cc < 4; ++cc) {
            float h  = __builtin_amdgcn_fmed3f(hv[cc], EPSV, ONE_M_EPS); // clip
            float xa = fmaf(h, sgn, off);                 // h or 1-h (branchless)
            unsigned act = (mask >> (col0 + cc)) & 1u;    // active-window bit
            float x  = act ? xa : 1.0f;                   // log(1) = 0 when inactive
            acc -= __logf(x);
        }
    }

    // Tail rows (n_rows % 4): one wave per row, lane = column.
    for (int r = n_groups * 4 + gwave; r < n_rows; r += nwaves) {
        float h = preds[(size_t)r * 32 + lane];
        int2 te = tgt2[r];
        unsigned mask; float sgn, off;
        row_params(te.x, te.y, mask, sgn, off);
        h = __builtin_amdgcn_fmed3f(h, EPSV, ONE_M_EPS);
        float xa = fmaf(h, sgn, off);
        unsigned act = (mask >> lane) & 1u;
        float x = act ? xa : 1.0f;
        acc -= __logf(x);
    }

    // Wave32 reduction via f32 WMMA (EXEC is all-ones here: loops are wave-uniform).
    float wsum = wave_sum32_wmma(acc);

    __shared__ float lds[8];
    if (lane == 0) lds[waveInBlk] = wsum;
    __syncthreads();
    if (threadIdx.x == 0) {
        float bsum = 0.0f;
#pragma unroll
        for (int i = 0; i < 8; ++i) bsum += lds[i];
        partial[blockIdx.x] = bsum;  // each block owns its slot: no init needed
    }
}

__global__ __launch_bounds__(256) void survloss_final_kernel(
    const float* __restrict__ partial, int n_parts,
    float* __restrict__ out, int n_rows) {
    __shared__ double lds[256];
    double s = 0.0;
    for (int i = threadIdx.x; i < n_parts; i += 256) s += (double)partial[i];
    lds[threadIdx.x] = s;
    __syncthreads();
    if (threadIdx.x == 0) {
        double tot = 0.0;
        for (int i = 0; i < 256; ++i) tot += lds[i];
        out[0] = (float)(tot / (double)n_rows);
    }
}

extern "C" void kernel_launch(void* const* d_in, const int* in_sizes, int n_in,
                              void* d_out, int out_size, void* d_ws, size_t ws_size,
                              hipStream_t stream) {
    const float* preds   = (const float*)d_in[0];
    const int*   targets = (const int*)d_in[1];
    float*       out     = (float*)d_out;
    float*       partial = (float*)d_ws;

    const int n_rows   = in_sizes[0] / 32;   // N (T = 32 fixed by reference)
    const int n_groups = n_rows / 4;         // 4-row groups (float4 per lane)

    int n_blocks = 2048;                     // ~16K waves: enough MLP for 23.3 TB/s
    size_t max_parts = ws_size / sizeof(float);
    if (max_parts > 0 && (size_t)n_blocks > max_parts)
        n_blocks = (int)max_parts;
    if (n_blocks < 1) n_blocks = 1;

    survloss_partial_kernel<<<n_blocks, 256, 0, stream>>>(
        preds, targets, partial, n_rows, n_groups);
    survloss_final_kernel<<<1, 256, 0, stream>>>(
        partial, n_blocks, out, n_rows);
}